// Gemma4DecoderWrapper_8564164788529
// MI455X (gfx1250) — compile-verified
//
#include <hip/hip_runtime.h>
#include <hip/hip_bf16.h>
#include <math.h>

// ---------------- model constants (match reference) ----------------
#define L_   16
#define D_   1024
#define H_   8
#define KV_  2
#define HDS_ 256
#define HDF_ 512
#define HDMAX_ 512
#define FF_  4096
#define P_   256
#define V_   32000
#define T_   1024
#define EPS_ 1e-6f
#define SOFTCAP_ 30.0f

// padded LDS row strides (avoid bank conflicts between the 4 live rows)
#define SPS_ 1028   // prob row stride (>= 1024)
#define QSP_ 516    // q row stride (>= 512)

typedef __attribute__((ext_vector_type(2))) float v2f;
typedef __attribute__((ext_vector_type(8))) float v8f;

// ---------------- helpers ----------------
__device__ __forceinline__ float block_reduce_sum(float v) {
    __shared__ float red[8];
    __shared__ float total;
    const int lane = threadIdx.x & 31;
    const int wid  = threadIdx.x >> 5;
    #pragma unroll
    for (int o = 16; o > 0; o >>= 1) v += __shfl_down(v, o, 32);
    if (lane == 0) red[wid] = v;
    __syncthreads();
    const int nw = blockDim.x >> 5;
    float s = (threadIdx.x < nw) ? red[threadIdx.x] : 0.f;
    if (wid == 0) {
        #pragma unroll
        for (int o = 4; o > 0; o >>= 1) s += __shfl_down(s, o, 32);
        if (lane == 0) total = s;
    }
    __syncthreads();
    return total;
}

__device__ __forceinline__ float gelu_tanh(float x) {
    const float k = 0.7978845608028654f; // sqrt(2/pi)
    return 0.5f * x * (1.f + tanhf(k * (x + 0.044715f * x * x * x)));
}

// ---------------- RMS norm (optionally +resid, *scalar) ----------------
// block b handles vector src + b*n -> dst + b*n
__global__ void __launch_bounds__(256)
rms_kernel(const float* __restrict__ src, const float* __restrict__ w,
           const float* __restrict__ resid, const float* __restrict__ scalar_p,
           float* __restrict__ dst, int n) {
    src += (size_t)blockIdx.x * n;
    dst += (size_t)blockIdx.x * n;
    float ss = 0.f;
    for (int i = threadIdx.x; i < n; i += blockDim.x) { float v = src[i]; ss += v * v; }
    ss = block_reduce_sum(ss);
    const float inv = rsqrtf(ss / (float)n + EPS_);
    const float sc  = scalar_p ? *scalar_p : 1.f;
    for (int i = threadIdx.x; i < n; i += blockDim.x) {
        float g = w ? (1.f + w[i]) : 1.f;
        float v = src[i] * inv * g;
        if (resid) v += resid[i];
        dst[i] = v * sc;
    }
}

// ---------------- matvec: y[M] = W[M x Dd] @ x, row stride ldw ----------------
// one wave per row, 8 rows per block, coalesced float4 streaming (bandwidth bound)
__global__ void __launch_bounds__(256)
matvec_kernel(const float* __restrict__ W, const float* __restrict__ x,
              float* __restrict__ y, int M, int Dd, int ldw) {
    const int row  = blockIdx.x * 8 + (threadIdx.x >> 5);
    const int lane = threadIdx.x & 31;
    if (row >= M) return;
    const float4* __restrict__ wr = (const float4*)(W + (size_t)row * (size_t)ldw);
    const float4* __restrict__ xv = (const float4*)x;
    float acc = 0.f;
    const int n4 = Dd >> 2;
    for (int c = lane; c < n4; c += 32) {
        float4 a = wr[c];
        float4 b = xv[c];
        acc += a.x * b.x + a.y * b.y + a.z * b.z + a.w * b.w;
    }
    #pragma unroll
    for (int o = 16; o > 0; o >>= 1) acc += __shfl_down(acc, o, 32);
    if (lane == 0) y[row] = acc;
}

// ---------------- per-head RMS norm + rotate-half RoPE (in place) ----------------
__global__ void __launch_bounds__(256)
qknorm_rope_kernel(float* __restrict__ buf, const float* __restrict__ normw,
                   const float* __restrict__ cosT, const float* __restrict__ sinT,
                   const int* __restrict__ pos_p, int hd) {
    float* v = buf + (size_t)blockIdx.x * hd;
    const int pos = *pos_p;
    const float* cr = cosT + (size_t)pos * hd;
    const float* sr = sinT + (size_t)pos * hd;
    float ss = 0.f;
    for (int i = threadIdx.x; i < hd; i += blockDim.x) { float a = v[i]; ss += a * a; }
    ss = block_reduce_sum(ss);
    const float inv = rsqrtf(ss / (float)hd + EPS_);
    const int half = hd >> 1;
    for (int i = threadIdx.x; i < half; i += blockDim.x) {
        float a = v[i]        * inv * (1.f + normw[i]);
        float b = v[i + half] * inv * (1.f + normw[i + half]);
        v[i]        = a * cr[i]        - b * sr[i];
        v[i + half] = b * cr[i + half] + a * sr[i + half];
    }
}

// ---------------- elementwise: out[i] = gelu_tanh(a[i]) * b[i] ----------------
__global__ void __launch_bounds__(256)
gelu_mul_kernel(const float* __restrict__ a, const float* __restrict__ b,
                float* __restrict__ out, int n) {
    int i = blockIdx.x * blockDim.x + threadIdx.x;
    if (i < n) out[i] = gelu_tanh(a[i]) * b[i];
}

// ---------------- attention (WMMA f32 16x16x4, branchless inner loops) ----------------
// grid.x = KV groups. Each group: 4 query heads, keys t in [0, pos].
// K/V row = cache for t < pos, freshly computed knew/vnew row for t == pos.
// q rows and probs staged in LDS; row 4 of each LDS array is all-zero so that
// inactive WMMA A-rows (heads 4..15) load zeros with NO divergent control flow.
__global__ void __launch_bounds__(256)
attn_kernel(const float* __restrict__ qbuf,   // [H][hd]
            const float* __restrict__ kc,     // kv_cache K, layer base: [KV][T][HDMAX]
            const float* __restrict__ vc,     // kv_cache V, layer base
            const float* __restrict__ knew,   // [KV][hd] contiguous
            const float* __restrict__ vnew,   // [KV][hd]
            const int* __restrict__ pos_p,
            float* __restrict__ attnbuf,      // [H][hd]
            int hd) {
    __shared__ float sprob[5 * SPS_];
    __shared__ float qs[5 * QSP_];
    const int g    = blockIdx.x;
    const int tid  = threadIdx.x;
    const int lane = tid & 31;
    const int wid  = tid >> 5;
    const int col  = lane & 15;     // A-row index for loads / C-column index
    const int hi   = lane >> 4;     // lane-half
    const int pos    = *pos_p;
    const int ntiles = (pos + 16) >> 4;   // covers t = 0..pos
    const int n16    = ntiles << 4;

    // stage q rows (+ zero row 4) into LDS; zero the prob pad row
    for (int i = tid; i < 5 * QSP_; i += 256) {
        const int r = i / QSP_, c = i - r * QSP_;
        qs[i] = (r < 4 && c < hd) ? qbuf[(size_t)(g * 4 + r) * hd + c] : 0.f;
    }
    for (int i = tid; i < SPS_; i += 256) sprob[4 * SPS_ + i] = 0.f;
    __syncthreads();

    const int arow = (col < 4) ? col : 4;          // zero row for inactive A rows
    const float* __restrict__ qrow = qs + arow * QSP_;

    // ---- phase 1: scores[h][t] = q[h] . K[t]  (one 16-key tile per wave) ----
    for (int it = wid; it < ntiles; it += 8) {
        const int t0 = it << 4;
        const int tB = t0 + col;
        const float* __restrict__ kp =
            (tB < pos) ? (kc + ((size_t)g * T_ + tB) * HDMAX_)
                       : (knew + (size_t)g * hd);
        v8f c;
        #pragma unroll
        for (int j = 0; j < 8; ++j) c[j] = 0.f;
        for (int kk = 0; kk < hd; kk += 4) {
            const int k0 = kk + hi * 2;
            v2f a, b;
            a[0] = qrow[k0];                  // A[m][2*hi+0] (ds_load_b64)
            a[1] = qrow[k0 + 1];
            b[0] = kp[k0];                    // B[2*hi+0][col] (global_load_b64)
            b[1] = kp[k0 + 1];
            c = __builtin_amdgcn_wmma_f32_16x16x4_f32(false, a, false, b,
                                                      (short)0, c, false, false);
        }
        #pragma unroll
        for (int j = 0; j < 8; ++j) {
            const int row = j + hi * 8;       // C row = head (local)
            if (row < 4) {
                const int t = t0 + col;
                sprob[row * SPS_ + t] = (t <= pos) ? c[j] : -1e30f;
            }
        }
    }
    __syncthreads();

    // ---- phase 2: softmax per head (one wave per head, no scaling per ref) ----
    if (wid < 4) {
        float* srow = sprob + wid * SPS_;
        float mx = -1e30f;
        for (int t = lane; t < n16; t += 32) mx = fmaxf(mx, srow[t]);
        #pragma unroll
        for (int o = 16; o > 0; o >>= 1) mx = fmaxf(mx, __shfl_xor(mx, o, 32));
        float sum = 0.f;
        for (int t = lane; t < n16; t += 32) {
            float e = expf(srow[t] - mx);
            srow[t] = e;
            sum += e;
        }
        #pragma unroll
        for (int o = 16; o > 0; o >>= 1) sum += __shfl_xor(sum, o, 32);
        const float invs = 1.f / sum;
        for (int t = lane; t < n16; t += 32) srow[t] *= invs;
    }
    __syncthreads();

    // ---- phase 3: attn[h][d] = sum_t p[h][t] * V[t][d] (16-wide d tiles) ----
    const float* __restrict__ prow = sprob + arow * SPS_;  // zero row if inactive
    const int ndt   = hd >> 4;
    const int tmain = pos & ~3;   // every row in [0,tmain) is strictly < pos
    for (int dt = wid; dt < ndt; dt += 8) {
        const int d0 = dt << 4;
        v8f c;
        #pragma unroll
        for (int j = 0; j < 8; ++j) c[j] = 0.f;
        // main loop: pure cache reads, zero selects
        const float* __restrict__ vb = vc + (size_t)g * T_ * HDMAX_ + d0 + col;
        int tt = 0;
        for (; tt < tmain; tt += 4) {
            const int k0 = tt + hi * 2;
            v2f a, b;
            a[0] = prow[k0];
            a[1] = prow[k0 + 1];
            b[0] = vb[(size_t)k0 * HDMAX_];
            b[1] = vb[(size_t)(k0 + 1) * HDMAX_];
            c = __builtin_amdgcn_wmma_f32_16x16x4_f32(false, a, false, b,
                                                      (short)0, c, false, false);
        }
        // tail: covers t == pos (fresh v) and masked padding rows (p == 0)
        for (; tt < n16; tt += 4) {
            const int k0 = tt + hi * 2;
            v2f a, b;
            a[0] = prow[k0];
            a[1] = prow[k0 + 1];
            const float* vp0 = (k0     < pos) ? (vc + ((size_t)g * T_ + k0)     * HDMAX_)
                                              : (vnew + (size_t)g * hd);
            const float* vp1 = (k0 + 1 < pos) ? (vc + ((size_t)g * T_ + k0 + 1) * HDMAX_)
                                              : (vnew + (size_t)g * hd);
            b[0] = vp0[d0 + col];
            b[1] = vp1[d0 + col];
            c = __builtin_amdgcn_wmma_f32_16x16x4_f32(false, a, false, b,
                                                      (short)0, c, false, false);
        }
        #pragma unroll
        for (int j = 0; j < 8; ++j) {
            const int row = j + hi * 8;
            if (row < 4) attnbuf[(size_t)(g * 4 + row) * hd + d0 + col] = c[j];
        }
    }
}

// ---------------- final argmax + tanh softcap ----------------
__global__ void __launch_bounds__(256)
argmax_kernel(const float* __restrict__ logits, int n, void* __restrict__ out) {
    __shared__ float sv[256];
    __shared__ int   si[256];
    float best = -1e30f;
    int bi = 0;
    for (int i = threadIdx.x; i < n; i += 256) {
        float v = logits[i];
        if (v > best) { best = v; bi = i; }
    }
    sv[threadIdx.x] = best;
    si[threadIdx.x] = bi;
    __syncthreads();
    for (int s = 128; s > 0; s >>= 1) {
        if (threadIdx.x < s) {
            float ov = sv[threadIdx.x + s];
            int   oi = si[threadIdx.x + s];
            if (ov > sv[threadIdx.x] || (ov == sv[threadIdx.x] && oi < si[threadIdx.x])) {
                sv[threadIdx.x] = ov;
                si[threadIdx.x] = oi;
            }
        }
        __syncthreads();
    }
    if (threadIdx.x == 0) {
        ((int*)out)[0]   = si[0];
        ((float*)out)[1] = tanhf(sv[0] / SOFTCAP_) * SOFTCAP_;
    }
}

// ---------------- host orchestration ----------------
extern "C" void kernel_launch(void* const* d_in, const int* in_sizes, int n_in,
                              void* d_out, int out_size, void* d_ws, size_t ws_size,
                              hipStream_t stream) {
    (void)in_sizes; (void)n_in; (void)out_size; (void)ws_size;
    const float* inputs_embeds = (const float*)d_in[0];
    const float* pli     = (const float*)d_in[1];
    const int*   pos_p   = (const int*)d_in[2];
    // d_in[3] causal_mask, d_in[4] update_mask: semantics realized via pos
    const float* kv      = (const float*)d_in[5];
    const float* cs      = (const float*)d_in[6];
    const float* ssin    = (const float*)d_in[7];
    const float* cf      = (const float*)d_in[8];
    const float* sfull   = (const float*)d_in[9];
    const float* q_w     = (const float*)d_in[10];
    const float* k_w     = (const float*)d_in[11];
    const float* v_w     = (const float*)d_in[12];
    const float* o_w     = (const float*)d_in[13];
    const float* q_nw    = (const float*)d_in[14];
    const float* k_nw    = (const float*)d_in[15];
    const float* ln_in   = (const float*)d_in[16];
    const float* ln_pa   = (const float*)d_in[17];
    const float* ln_pf   = (const float*)d_in[18];
    const float* ln_ff   = (const float*)d_in[19];
    const float* ln_pl   = (const float*)d_in[20];
    const float* gate_w  = (const float*)d_in[21];
    const float* up_w    = (const float*)d_in[22];
    const float* down_w  = (const float*)d_in[23];
    const float* plg_w   = (const float*)d_in[24];
    const float* plp_w   = (const float*)d_in[25];
    const float* lscal   = (const float*)d_in[26];
    const float* fin_nw  = (const float*)d_in[27];
    const float* lm_head = (const float*)d_in[28];

    // workspace carve-up (floats)
    float* ws = (float*)d_ws;
    float* wh     = ws;  ws += D_;                  // hidden state
    float* wx     = ws;  ws += D_;                  // normed activations
    float* wq     = ws;  ws += H_ * HDMAX_;         // q heads
    float* wknew  = ws;  ws += L_ * KV_ * HDMAX_;   // per-layer fresh k (rows of length hd)
    float* wvnew  = ws;  ws += L_ * KV_ * HDMAX_;   // per-layer fresh v
    float* wattn  = ws;  ws += H_ * HDMAX_;         // attention output heads
    float* wgate  = ws;  ws += FF_;
    float* wup    = ws;  ws += FF_;
    float* wact   = ws;  ws += FF_;
    float* wy     = ws;  ws += D_;                  // matvec target (attn_out / mlp_out / plp_out)
    float* wplg   = ws;  ws += P_;
    float* wg     = ws;  ws += P_;
    float* wlog   = ws;  ws += V_;

    hipMemcpyAsync(wh, inputs_embeds, D_ * sizeof(float),
                   hipMemcpyDeviceToDevice, stream);

    const size_t KVTH = (size_t)KV_ * T_ * HDMAX_;

    for (int l = 0; l < L_; ++l) {
        const bool is_full   = ((l % 5) == 4);
        const bool is_shared = (l >= L_ - 1);           // NUM_KV_SHARED = 1
        const int  hd        = is_full ? HDF_ : HDS_;
        const float* cosT = is_full ? cf    : cs;
        const float* sinT = is_full ? sfull : ssin;
        const int  src = is_shared ? (is_full ? 14 : 13) : l;

        // x = rms(h, ln_in[l])
        rms_kernel<<<1, 256, 0, stream>>>(wh, ln_in + (size_t)l * D_, nullptr, nullptr, wx, D_);

        // q = q_w[l,:H*hd] @ x ; per-head rms + rope
        matvec_kernel<<<(H_ * hd) / 8, 256, 0, stream>>>(
            q_w + (size_t)l * H_ * HDMAX_ * D_, wx, wq, H_ * hd, D_, D_);
        qknorm_rope_kernel<<<H_, 256, 0, stream>>>(
            wq, q_nw + (size_t)l * HDMAX_, cosT, sinT, pos_p, hd);

        float* knew_l = wknew + (size_t)l * KV_ * HDMAX_;
        float* vnew_l = wvnew + (size_t)l * KV_ * HDMAX_;
        if (!is_shared) {
            matvec_kernel<<<(KV_ * hd) / 8, 256, 0, stream>>>(
                k_w + (size_t)l * KV_ * HDMAX_ * D_, wx, knew_l, KV_ * hd, D_, D_);
            qknorm_rope_kernel<<<KV_, 256, 0, stream>>>(
                knew_l, k_nw + (size_t)l * HDMAX_, cosT, sinT, pos_p, hd);
            matvec_kernel<<<(KV_ * hd) / 8, 256, 0, stream>>>(
                v_w + (size_t)l * KV_ * HDMAX_ * D_, wx, vnew_l, KV_ * hd, D_, D_);
            // _vnorm: rms without weight, in place, per kv head
            rms_kernel<<<KV_, 256, 0, stream>>>(vnew_l, nullptr, nullptr, nullptr, vnew_l, hd);
        }

        // attention (reads cache rows t<pos of source layer, fresh kv at t==pos)
        const float* kc = kv + (size_t)src * KVTH;
        const float* vc = kv + (size_t)(L_ + src) * KVTH;
        const float* ksrc = wknew + (size_t)src * KV_ * HDMAX_;
        const float* vsrc = wvnew + (size_t)src * KV_ * HDMAX_;
        attn_kernel<<<KV_, 256, 0, stream>>>(wq, kc, vc, ksrc, vsrc, pos_p, wattn, hd);

        // o-proj + post-attn rms + residual
        matvec_kernel<<<D_ / 8, 256, 0, stream>>>(
            o_w + (size_t)l * D_ * H_ * HDMAX_, wattn, wy, D_, H_ * hd, H_ * HDMAX_);
        rms_kernel<<<1, 256, 0, stream>>>(wy, ln_pa + (size_t)l * D_, wh, nullptr, wh, D_);

        // MLP
        rms_kernel<<<1, 256, 0, stream>>>(wh, ln_pf + (size_t)l * D_, nullptr, nullptr, wx, D_);
        matvec_kernel<<<FF_ / 8, 256, 0, stream>>>(
            gate_w + (size_t)l * FF_ * D_, wx, wgate, FF_, D_, D_);
        matvec_kernel<<<FF_ / 8, 256, 0, stream>>>(
            up_w + (size_t)l * FF_ * D_, wx, wup, FF_, D_, D_);
        gelu_mul_kernel<<<FF_ / 256, 256, 0, stream>>>(wgate, wup, wact, FF_);
        matvec_kernel<<<D_ / 8, 256, 0, stream>>>(
            down_w + (size_t)l * D_ * FF_, wact, wy, D_, FF_, FF_);
        rms_kernel<<<1, 256, 0, stream>>>(wy, ln_ff + (size_t)l * D_, wh, nullptr, wh, D_);

        // per-layer projection branch
        matvec_kernel<<<P_ / 8, 256, 0, stream>>>(
            plg_w + (size_t)l * P_ * D_, wh, wplg, P_, D_, D_);
        gelu_mul_kernel<<<P_ / 256, 256, 0, stream>>>(wplg, pli + (size_t)l * P_, wg, P_);
        matvec_kernel<<<D_ / 8, 256, 0, stream>>>(
            plp_w + (size_t)l * D_ * P_, wg, wy, D_, P_, P_);
        rms_kernel<<<1, 256, 0, stream>>>(wy, ln_pl + (size_t)l * D_, wh, lscal + l, wh, D_);
    }

    // final norm + lm_head + softcapped argmax
    rms_kernel<<<1, 256, 0, stream>>>(wh, fin_nw, nullptr, nullptr, wx, D_);
    matvec_kernel<<<V_ / 8, 256, 0, stream>>>(lm_head, wx, wlog, V_, D_, D_);
    argmax_kernel<<<1, 256, 0, stream>>>(wlog, V_, d_out);
}